// Flashback_57818849738853
// MI455X (gfx1250) — compile-verified
//
#include <hip/hip_runtime.h>
#include <math.h>

#define V_  30000
#define H_  128
#define S_  32
#define U_  128
#define E_  300000
#define SU_ (S_*U_)
#define K2_ 256

typedef __attribute__((ext_vector_type(2))) float v2f;
typedef __attribute__((ext_vector_type(8))) float v8f;

__device__ __forceinline__ v8f wmma_f32(v2f a, v2f b, v8f c) {
  // V_WMMA_F32_16X16X4_F32: D = A(16x4) * B(4x16) + C(16x16), fp32 exact
  return __builtin_amdgcn_wmma_f32_16x16x4_f32(
      false, a, false, b, (short)0, c, false, false);
}

// ---- init: agg=0, deg=1 (self loop), flags=0 -------------------------------
__global__ void k_zero(float* agg, float* deg, int* flags) {
  long i = (long)blockIdx.x * 256 + threadIdx.x;
  if (i < (long)V_ * H_) agg[i] = 0.f;
  if (i < V_) { deg[i] = 1.f; flags[i] = 0; }
}

// ---- mark rows of encoder_weight that x actually gathers -------------------
__global__ void k_flag(const int* x, int* flags) {
  int i = blockIdx.x * 256 + threadIdx.x;
  if (i < SU_) flags[x[i]] = 1;
}

// ---- deg[row] += 1 per edge ------------------------------------------------
__global__ void k_deg(const int* rows, float* deg) {
  int i = blockIdx.x * 256 + threadIdx.x;
  if (i < E_) atomicAdd(&deg[rows[i]], 1.f);
}

// ---- agg[row] += enc_emb[col], filtered to needed rows ---------------------
__global__ void k_agg(const int* rows, const int* cols, const int* flags,
                      const float* enc, float* agg) {
  long tid = (long)blockIdx.x * 256 + threadIdx.x;   // E_*32 threads
  int e = (int)(tid >> 5);
  int h0 = ((int)tid & 31) * 4;
  int r = rows[e];
  if (flags[r]) {
    int c = cols[e];
    #pragma unroll
    for (int q = 0; q < 4; ++q)
      atomicAdd(&agg[(long)r * H_ + h0 + q], enc[(long)c * H_ + h0 + q]);
  }
}

// ---- x_emb = (agg[x] + enc[x]) / deg[x] ------------------------------------
__global__ void k_xemb(const int* x, const float* agg, const float* enc,
                       const float* deg, float* xemb) {
  int i = blockIdx.x * 256 + threadIdx.x;            // SU_*H_
  int row = i / H_, hh = i % H_;
  int xi = x[row];
  xemb[i] = (agg[(long)xi * H_ + hh] + enc[(long)xi * H_ + hh]) / deg[xi];
}

// ---- Brnn[k][n] = (k<128 ? W_ih[n][k] : W_hh[n][k-128]) --------------------
__global__ void k_brnn(const float* Wih, const float* Whh, float* B) {
  int i = blockIdx.x * 256 + threadIdx.x;            // 256*128
  int k = i / 128, n = i % 128;
  B[i] = (k < H_) ? Wih[n * H_ + k] : Whh[n * H_ + (k - H_)];
}

// ---- one RNN step: h_out = tanh([x_s|h_prev](128x256) @ Brnn(256x128) + b) -
__global__ void k_rnn(const float* __restrict__ x_s,
                      const float* __restrict__ h_prev,
                      const float* __restrict__ B,
                      const float* __restrict__ brnn,
                      float* __restrict__ h_out) {
  int wid  = blockIdx.x * 8 + (threadIdx.x >> 5);    // 64 waves = 8x8 tiles
  int lane = threadIdx.x & 31;
  int mt = wid >> 3, nt = wid & 7;
  int m0 = mt * 16, n0 = nt * 16;
  int ml = lane & 15, hi = lane >> 4;
  v8f acc = {};
  for (int k0 = 0; k0 < K2_; k0 += 4) {
    int kk = k0 + 2 * hi;
    const float* A = (k0 < H_) ? x_s : h_prev;
    int ka = (k0 < H_) ? kk : (kk - H_);
    v2f a; a.x = A[(m0 + ml) * H_ + ka]; a.y = A[(m0 + ml) * H_ + ka + 1];
    v2f b; b.x = B[kk * 128 + n0 + ml]; b.y = B[(kk + 1) * 128 + n0 + ml];
    acc = wmma_f32(a, b, acc);
  }
  #pragma unroll
  for (int r = 0; r < 8; ++r) {
    int row = m0 + r + 8 * hi, col = n0 + ml;
    h_out[row * H_ + col] = tanhf(acc[r] + brnn[col]);
  }
}

// ---- p_u gather + broadcast into right half of Amat ------------------------
__global__ void k_pu(const int* act, const float* uemb, float* pu, float* Amat) {
  int i = blockIdx.x * 256 + threadIdx.x;            // SU_*H_
  int row = i / H_, hh = i % H_;
  int s = row / U_, u = row % U_;
  float v = uemb[(long)act[u] * H_ + hh];
  Amat[(long)row * K2_ + H_ + hh] = v;
  if (s == 0) pu[u * H_ + hh] = v;
}

// ---- pp[u][j] = tanh(p_u @ proj_W + b)[j] + pref[j] ------------------------
__global__ void k_pp(const float* pu, const float* pW, const float* pb,
                     const float* pref, float* pp) {
  int i = blockIdx.x * 256 + threadIdx.x;            // U_*256
  int u = i >> 8, j = i & 255;
  float acc = pb[j];
  for (int k = 0; k < H_; ++k) acc += pu[u * H_ + k] * pW[k * K2_ + j];
  pp[i] = tanhf(acc) + pref[j];
}

// ---- sim[u][s] = sigmoid(sum_j pp[u][j]*tanh((x_emb@proj_W+b)[s,u,j])) -----
__global__ void k_sim(const float* xemb, const float* pW, const float* pb,
                      const float* pp, float* sim) {
  int wid  = blockIdx.x * 8 + (threadIdx.x >> 5);    // SU_ waves
  int lane = threadIdx.x & 31;
  int s = wid / U_, u = wid % U_;
  const float* xr = xemb + (long)(s * U_ + u) * H_;
  float tot = 0.f;
  #pragma unroll
  for (int j0 = 0; j0 < 8; ++j0) {
    int j = lane + 32 * j0;
    float acc = pb[j];
    for (int k = 0; k < H_; ++k) acc += xr[k] * pW[k * K2_ + j];
    tot += pp[u * K2_ + j] * tanhf(acc);
  }
  for (int m = 16; m; m >>= 1) tot += __shfl_xor(tot, m, 32);
  if (lane == 0) sim[u * S_ + s] = 1.f / (1.f + expf(-tot));
}

// ---- causal spatio-temporal weights + weighted average into Amat[:, :128] --
__global__ void k_outw(const float* t, const float* sp, const float* sim,
                       const float* rout, float* Amat) {
  __shared__ float wbuf[8][33];
  int wslot = threadIdx.x >> 5, lane = threadIdx.x & 31;
  int wid = blockIdx.x * 8 + wslot;                  // SU_ waves
  int i = wid / U_, u = wid % U_;
  int j = lane;
  float dt = t[i * U_ + u] - t[j * U_ + u];
  float dx = sp[(i * U_ + u) * 2]     - sp[(j * U_ + u) * 2];
  float dy = sp[(i * U_ + u) * 2 + 1] - sp[(j * U_ + u) * 2 + 1];
  float dsn = sqrtf(dx * dx + dy * dy);
  float a = 0.5f * (cosf(dt * 7.27220521664304e-5f) + 1.f)
                 * expf(-(dt / 86400.f) * 0.1f);
  float b = expf(-dsn * 100.f);
  float wj = (a * b + 1e-10f) * sim[u * S_ + j];
  if (j > i) wj = 0.f;
  float sw = wj;
  for (int m = 16; m; m >>= 1) sw += __shfl_xor(sw, m, 32);
  wbuf[wslot][lane] = wj;
  __syncthreads();
  float inv = 1.f / sw;
  for (int h = lane; h < H_; h += 32) {
    float acc = 0.f;
    for (int jj = 0; jj <= i; ++jj)
      acc += wbuf[wslot][jj] * rout[(jj * U_ + u) * H_ + h];
    Amat[(long)(i * U_ + u) * K2_ + h] = acc * inv;
  }
}

// ---- y = Amat(4096x256) @ fc_W(256x30000) + fc_b : store-BW bound ----------
__global__ void k_gemm(const float* __restrict__ Amat,
                       const float* __restrict__ fcW,
                       const float* __restrict__ fcb,
                       float* __restrict__ out) {
  int nq = blockIdx.x % 625;                         // 625*48 = 30000 cols
  int mg = blockIdx.x / 625;                         // 32 groups * 8 waves
  int wv = threadIdx.x >> 5, lane = threadIdx.x & 31;
  int mt = mg * 8 + wv;
  int m0 = mt * 16, n0 = nq * 48;
  int ml = lane & 15, hi = lane >> 4;
  const float* Ar = Amat + (long)(m0 + ml) * K2_;
  v8f c0 = {}, c1 = {}, c2 = {};
  for (int k0 = 0; k0 < K2_; k0 += 4) {
    int kk = k0 + 2 * hi;
    v2f a; a.x = Ar[kk]; a.y = Ar[kk + 1];
    const float* B0 = fcW + (size_t)kk * V_ + n0 + ml;
    const float* B1 = fcW + (size_t)(kk + 1) * V_ + n0 + ml;
    v2f b;
    b.x = B0[0];  b.y = B1[0];  c0 = wmma_f32(a, b, c0);
    b.x = B0[16]; b.y = B1[16]; c1 = wmma_f32(a, b, c1);
    b.x = B0[32]; b.y = B1[32]; c2 = wmma_f32(a, b, c2);
  }
  #pragma unroll
  for (int r = 0; r < 8; ++r) {
    int row = m0 + r + 8 * hi;
    size_t base = (size_t)row * V_ + n0 + ml;
    out[base]      = c0[r] + fcb[n0 + ml];
    out[base + 16] = c1[r] + fcb[n0 + 16 + ml];
    out[base + 32] = c2[r] + fcb[n0 + 32 + ml];
  }
}

// ---- h_last = rnn_out[S-1] appended after y --------------------------------
__global__ void k_hlast(const float* rout, float* out) {
  int i = blockIdx.x * 256 + threadIdx.x;            // U_*H_
  if (i < U_ * H_) out[(size_t)SU_ * V_ + i] = rout[(S_ - 1) * U_ * H_ + i];
}

extern "C" void kernel_launch(void* const* d_in, const int* in_sizes, int n_in,
                              void* d_out, int out_size, void* d_ws, size_t ws_size,
                              hipStream_t stream) {
  (void)in_sizes; (void)n_in; (void)out_size; (void)ws_size;
  const int*   x     = (const int*)  d_in[0];
  const float* t     = (const float*)d_in[1];
  const float* sp    = (const float*)d_in[3];
  const float* h0    = (const float*)d_in[7];
  const int*   act   = (const int*)  d_in[8];
  const int*   grow  = (const int*)  d_in[9];
  const int*   gcol  = (const int*)  d_in[10];
  const float* enc   = (const float*)d_in[11];
  const float* uemb  = (const float*)d_in[12];
  const float* pref  = (const float*)d_in[13];
  const float* pW    = (const float*)d_in[14];
  const float* pb    = (const float*)d_in[15];
  const float* Wih   = (const float*)d_in[16];
  const float* Whh   = (const float*)d_in[17];
  const float* brnn  = (const float*)d_in[18];
  const float* fcW   = (const float*)d_in[19];
  const float* fcb   = (const float*)d_in[20];
  float* out = (float*)d_out;

  // workspace carve-up (floats, 256B-aligned chunks)
  float* ws = (float*)d_ws;
  size_t o = 0;
  auto alloc = [&](size_t n) { size_t r = o; o += (n + 63) & ~(size_t)63; return r; };
  float* deg   = ws + alloc(V_);
  float* agg   = ws + alloc((size_t)V_ * H_);
  int*   flags = (int*)(ws + alloc(V_));
  float* xemb  = ws + alloc((size_t)SU_ * H_);
  float* Brnn  = ws + alloc(256 * 128);
  float* rout  = ws + alloc((size_t)S_ * U_ * H_);
  float* pu    = ws + alloc((size_t)U_ * H_);
  float* pp    = ws + alloc((size_t)U_ * K2_);
  float* sim   = ws + alloc((size_t)U_ * S_);
  float* Amat  = ws + alloc((size_t)SU_ * K2_);

  k_zero<<<((size_t)V_ * H_ + 255) / 256, 256, 0, stream>>>(agg, deg, flags);
  k_flag<<<(SU_ + 255) / 256, 256, 0, stream>>>(x, flags);
  k_deg<<<(E_ + 255) / 256, 256, 0, stream>>>(grow, deg);
  k_agg<<<((size_t)E_ * 32) / 256, 256, 0, stream>>>(grow, gcol, flags, enc, agg);
  k_xemb<<<(SU_ * H_) / 256, 256, 0, stream>>>(x, agg, enc, deg, xemb);
  k_brnn<<<(256 * 128) / 256, 256, 0, stream>>>(Wih, Whh, Brnn);

  // sequential RNN scan: 32 small WMMA kernels, state chained through rout
  for (int s = 0; s < S_; ++s) {
    const float* hprev = (s == 0) ? h0 : (rout + (size_t)(s - 1) * U_ * H_);
    k_rnn<<<8, 256, 0, stream>>>(xemb + (size_t)s * U_ * H_, hprev, Brnn, brnn,
                                 rout + (size_t)s * U_ * H_);
  }

  k_pu<<<(SU_ * H_) / 256, 256, 0, stream>>>(act, uemb, pu, Amat);
  k_pp<<<(U_ * K2_) / 256, 256, 0, stream>>>(pu, pW, pb, pref, pp);
  k_sim<<<SU_ / 8, 256, 0, stream>>>(xemb, pW, pb, pp, sim);
  k_outw<<<SU_ / 8, 256, 0, stream>>>(t, sp, sim, rout, Amat);

  // dominant kernel: 491 MB output write, fp32 WMMA
  k_gemm<<<625 * 32, 256, 0, stream>>>(Amat, fcW, fcb, out);
  k_hlast<<<(U_ * H_ + 255) / 256, 256, 0, stream>>>(rout, out);
}